// ORCDF_77249281786067
// MI455X (gfx1250) — compile-verified
//
#include <hip/hip_runtime.h>
#include <math.h>

typedef __attribute__((ext_vector_type(2))) float v2f;
typedef __attribute__((ext_vector_type(8))) float v8f;

#define DIM 64

// ---------------------------------------------------------------------------
// COO SpMM: out[r] += val * src[c]  (or transposed: out[c] += val * src[r])
// One wave per edge: 32 lanes x float2 = 64 dims. Atomic f32 accumulate.
// Prefetch next edge's source row (gfx1250 global_prefetch_b8).
// ---------------------------------------------------------------------------
__global__ void spmm_edges(const int* __restrict__ rows, const int* __restrict__ cols,
                           const float* __restrict__ vals, const float* __restrict__ src,
                           float* __restrict__ out, int nnz, int transpose)
{
    const int lane   = threadIdx.x & 31;
    const int wid    = (blockIdx.x * blockDim.x + threadIdx.x) >> 5;
    const int nwaves = (gridDim.x * blockDim.x) >> 5;
    for (int e = wid; e < nnz; e += nwaves) {
        int r = rows[e];
        int c = cols[e];
        if (transpose) { int t = r; r = c; c = t; }
        const float v = vals[e];
        const float2* s = reinterpret_cast<const float2*>(src + (size_t)c * DIM);
        float2 x = s[lane];
        int en = e + nwaves;
        if (en < nnz) {
            int cn = transpose ? rows[en] : cols[en];
            __builtin_prefetch(src + (size_t)cn * DIM + lane * 2, 0, 0);
        }
        float* o = out + (size_t)r * DIM + lane * 2;
        atomicAdd(o,     v * x.x);
        atomicAdd(o + 1, v * x.y);
    }
}

__global__ void zero_buf(float* __restrict__ p, int n)
{
    int i = blockIdx.x * blockDim.x + threadIdx.x;
    if (i < n) p[i] = 0.0f;
}

__global__ void copy_buf(float* __restrict__ dst, const float* __restrict__ src, int n)
{
    int i = blockIdx.x * blockDim.x + threadIdx.x;
    if (i < n) dst[i] = src[i];
}

__global__ void axpy_acc(float* __restrict__ dst, const float* __restrict__ src, int n)
{
    int i = blockIdx.x * blockDim.x + threadIdx.x;
    if (i < n) dst[i] += src[i];
}

__global__ void scale_buf(float* __restrict__ p, float s, int n)
{
    int i = blockIdx.x * blockDim.x + threadIdx.x;
    if (i < n) p[i] *= s;
}

// ---------------------------------------------------------------------------
// Predict tail: per wave, 16 batch rows. inter (16x128) @ W (128x1) via
// chained V_WMMA_F32_16X16X4_F32 (K stepped by 4). B operand broadcasts W
// across all 16 columns, so every column of D equals the logits; lane 0 holds
// logits[0..7] in its 8 C VGPRs, lane 16 holds logits[8..15].
// A-operand layout (32-bit A 16x4): lanes 0-15 -> K=0,1; lanes 16-31 -> K=2,3.
// ---------------------------------------------------------------------------
__global__ void predict_wmma(const int* __restrict__ sids, const int* __restrict__ pids,
                             const float* __restrict__ s_sum, const float* __restrict__ p_fin,
                             const float* __restrict__ W, const float* __restrict__ bias,
                             float* __restrict__ out, int B)
{
    const int lane = threadIdx.x & 31;
    const int wave = (blockIdx.x * blockDim.x + threadIdx.x) >> 5;
    const int base = wave * 16;
    if (base >= B) return;                 // uniform per-wave: EXEC stays all-1s

    const int m  = lane & 15;              // batch row within tile
    const int hi = lane >> 4;              // K half select (0: K,K+1 ; 1: K+2,K+3)
    const int bidx = base + m;
    const int sid = sids[bidx];
    const int pid = pids[bidx];
    const float* srow = s_sum + (size_t)sid * DIM;
    const float* prow = p_fin + (size_t)pid * DIM - DIM;   // so index k>=64 reads k-64

    v8f c = {};
#pragma unroll
    for (int kb = 0; kb < 2 * DIM; kb += 4) {
        const int k = kb + 2 * hi;
        const bool is_s  = (k < DIM);                       // never straddles (k even)
        const float* row = is_s ? srow : prow;
        const float  sc  = is_s ? 0.25f : 1.0f;             // fold final_s = s_sum/4
        v2f a  = { sc * row[k], sc * row[k + 1] };
        v2f bm = { W[k], W[k + 1] };
        c = __builtin_amdgcn_wmma_f32_16x16x4_f32(
                /*neg_a=*/false, a, /*neg_b=*/false, bm,
                /*c_mod=*/(short)0, c, /*reuse_a=*/false, /*reuse_b=*/false);
    }

    if (m == 0) {                           // lanes 0 and 16 each own 8 logits
        const float bb = bias[0];
        float* o = out + base + 8 * hi;
#pragma unroll
        for (int v = 0; v < 8; ++v) {
            float l = c[v] + bb;
            o[v] = 1.0f / (1.0f + __expf(-l));
        }
    }
}

// ---------------------------------------------------------------------------
extern "C" void kernel_launch(void* const* d_in, const int* in_sizes, int n_in,
                              void* d_out, int out_size, void* d_ws, size_t ws_size,
                              hipStream_t stream)
{
    const int*   student_ids = (const int*)  d_in[0];
    const int*   problem_ids = (const int*)  d_in[1];
    const int*   a_rows      = (const int*)  d_in[2];
    const int*   a_cols      = (const int*)  d_in[3];
    const float* a_vals      = (const float*)d_in[4];
    const int*   ia_rows     = (const int*)  d_in[5];
    const int*   ia_cols     = (const int*)  d_in[6];
    const float* ia_vals     = (const float*)d_in[7];
    const int*   q_rows      = (const int*)  d_in[8];
    const int*   q_cols      = (const int*)  d_in[9];
    const float* q_vals      = (const float*)d_in[10];
    const float* s_emb       = (const float*)d_in[11];
    const float* p_emb       = (const float*)d_in[12];
    const float* k_emb       = (const float*)d_in[13];
    const float* W           = (const float*)d_in[14];
    const float* bvec        = (const float*)d_in[15];

    const int B    = in_sizes[0];
    const int nnz  = in_sizes[2];
    const int qnnz = in_sizes[8];
    const int S    = in_sizes[11] / DIM;
    const int P    = in_sizes[12] / DIM;

    const size_t sD = (size_t)S * DIM;
    const size_t pD = (size_t)P * DIM;

    float* ws    = (float*)d_ws;
    float* s_buf = ws;  ws += sD;     // layer-propagated student features
    float* s_sum = ws;  ws += sD;     // running sum over layers
    float* p_b0  = ws;  ws += pD;     // problem ping
    float* p_b1  = ws;  ws += pD;     // problem pong
    float* p_sum = ws;  ws += pD;     // running sum over layers
    (void)ws_size; (void)n_in; (void)out_size;

    const dim3 blk(256);
    const int sN = (int)sD, pN = (int)pD;
    const int gS = (sN + 255) / 256;
    const int gP = (pN + 255) / 256;
    const int spmm_blocks = 4096;     // 32768 waves grid-striding the edge list

    // s_sum = s_emb ; p_sum = p_emb
    copy_buf<<<gS, blk, 0, stream>>>(s_sum, s_emb, sN);
    copy_buf<<<gP, blk, 0, stream>>>(p_sum, p_emb, pN);

    const float* src_s = s_emb;
    const float* src_p = p_emb;
    float* pbufs[2] = { p_b0, p_b1 };

    for (int layer = 0; layer < 3; ++layer) {
        float* new_p = pbufs[layer & 1];
        // new_p = A^T * src_s + IA^T * src_s
        zero_buf<<<gP, blk, 0, stream>>>(new_p, pN);
        spmm_edges<<<spmm_blocks, blk, 0, stream>>>(a_rows,  a_cols,  a_vals,  src_s, new_p, nnz, 1);
        spmm_edges<<<spmm_blocks, blk, 0, stream>>>(ia_rows, ia_cols, ia_vals, src_s, new_p, nnz, 1);
        // new_s = A * src_p + IA * src_p   (src_s fully consumed above -> s_buf reusable)
        zero_buf<<<gS, blk, 0, stream>>>(s_buf, sN);
        spmm_edges<<<spmm_blocks, blk, 0, stream>>>(a_rows,  a_cols,  a_vals,  src_p, s_buf, nnz, 0);
        spmm_edges<<<spmm_blocks, blk, 0, stream>>>(ia_rows, ia_cols, ia_vals, src_p, s_buf, nnz, 0);
        // running sums
        axpy_acc<<<gS, blk, 0, stream>>>(s_sum, s_buf, sN);
        axpy_acc<<<gP, blk, 0, stream>>>(p_sum, new_p, pN);
        src_s = s_buf;
        src_p = new_p;
    }

    // final_p = p_sum/4 + Q @ k_emb   (final_s = s_sum/4 folded into predict)
    scale_buf<<<gP, blk, 0, stream>>>(p_sum, 0.25f, pN);
    spmm_edges<<<1024, blk, 0, stream>>>(q_rows, q_cols, q_vals, k_emb, p_sum, qnnz, 0);

    // dense predict tail: 16 batch rows per wave, 8 waves per block
    const int pred_blocks = (B + 127) / 128;
    predict_wmma<<<pred_blocks, blk, 0, stream>>>(student_ids, problem_ids,
                                                  s_sum, p_sum, W, bvec,
                                                  (float*)d_out, B);
}